// KroneckerLinear_29446295781765
// MI455X (gfx1250) — compile-verified
//
#include <hip/hip_runtime.h>

// CDNA5 / gfx1250, wave32. out[N,64] = x[N,64] @ kron(f1,f2)^T + bias, all fp32.
// Strategy: B = W^T built per-lane in registers from f1,f2 (Kronecker structure),
// 16-row x tile per wave, 4 x (16x16) f32 accumulators, V_WMMA_F32_16X16X4_F32.

typedef __attribute__((ext_vector_type(2))) float v2f;
typedef __attribute__((ext_vector_type(8))) float v8f;

#define WAVES_PER_BLOCK 8

__global__ __launch_bounds__(256) void kron_linear_wmma_f32(
    const float* __restrict__ x,
    const float* __restrict__ f1,
    const float* __restrict__ f2,
    const float* __restrict__ bias,
    float* __restrict__ out,
    int n_rows)
{
    const int lane   = threadIdx.x & 31;
    const int wid    = threadIdx.x >> 5;
    const int lane15 = lane & 15;
    const int hi     = lane >> 4;   // 0 for lanes 0-15, 1 for lanes 16-31

    // ---------------------------------------------------------------
    // Build all B fragments in registers.
    // For GEMM D = A*B with out = x @ W^T:  B[k][c] = W[c][k] = f1[c/8][k/8]*f2[c%8][k%8].
    // WMMA f32 16x16x4 B layout (4x16, 2 VGPRs): lane n<16 holds {B[k0][c],B[k0+1][c]}
    // with k0 = base, c = n; lanes 16-31 hold k0 = base+2, c = n-16.
    // ---------------------------------------------------------------
    v2f bfrag[16][4];
    #pragma unroll
    for (int kk = 0; kk < 16; ++kk) {
        const int k0 = 4 * kk + 2 * hi;          // even, so k0 and k0+1 share k/8
        #pragma unroll
        for (int t = 0; t < 4; ++t) {
            const int c  = 16 * t + lane15;
            const float a1 = f1[(c >> 3) * 8 + (k0 >> 3)];
            const float b0 = f2[(c & 7) * 8 + (k0 & 7)];
            const float b1 = f2[(c & 7) * 8 + ((k0 + 1) & 7)];
            bfrag[kk][t].x = a1 * b0;
            bfrag[kk][t].y = a1 * b1;
        }
    }

    // Bias per column tile (C/D layout: column = 16*t + lane15, same for all 8 acc elems)
    float bias_t[4];
    #pragma unroll
    for (int t = 0; t < 4; ++t) bias_t[t] = bias[16 * t + lane15];

    const int n_tiles     = n_rows >> 4;                       // 16 rows per tile
    const int gwave       = blockIdx.x * WAVES_PER_BLOCK + wid;
    const int wave_stride = gridDim.x * WAVES_PER_BLOCK;

    for (int tile = gwave; tile < n_tiles; tile += wave_stride) {
        const long r0 = (long)tile << 4;
        // A fragment source: lane reads x[r0 + lane15][4*kk + 2*hi .. +1]
        const float* xrow = x + (r0 + lane15) * 64 + 2 * hi;

        v8f acc[4];
        #pragma unroll
        for (int t = 0; t < 4; ++t) {
            #pragma unroll
            for (int v = 0; v < 8; ++v) acc[t][v] = bias_t[t];
        }

        #pragma unroll
        for (int kk = 0; kk < 16; ++kk) {
            const v2f a = *(const v2f*)(xrow + 4 * kk);
            #pragma unroll
            for (int t = 0; t < 4; ++t) {
                acc[t] = __builtin_amdgcn_wmma_f32_16x16x4_f32(
                    /*neg_a=*/false, a,
                    /*neg_b=*/false, bfrag[kk][t],
                    /*c_mod=*/(short)0, acc[t],
                    /*reuse_a=*/false, /*reuse_b=*/false);
            }
        }

        // C/D layout: acc[t][v] -> row = (hi ? 8 : 0) + v, col = 16*t + lane15
        float* obase = out + (r0 + hi * 8) * 64 + lane15;
        #pragma unroll
        for (int t = 0; t < 4; ++t) {
            #pragma unroll
            for (int v = 0; v < 8; ++v) {
                obase[(long)v * 64 + 16 * t] = acc[t][v];
            }
        }
    }
}

extern "C" void kernel_launch(void* const* d_in, const int* in_sizes, int n_in,
                              void* d_out, int out_size, void* d_ws, size_t ws_size,
                              hipStream_t stream) {
    const float* x  = (const float*)d_in[0];   // [N, 64]
    const float* f1 = (const float*)d_in[1];   // [8, 8]
    const float* f2 = (const float*)d_in[2];   // [8, 8]
    const float* b  = (const float*)d_in[3];   // [64]
    float* out = (float*)d_out;                // [N, 64]

    const int n_rows  = in_sizes[0] / 64;      // 2097152
    const int n_tiles = n_rows >> 4;
    int blocks = (n_tiles + WAVES_PER_BLOCK - 1) / WAVES_PER_BLOCK;
    if (blocks > 2048) blocks = 2048;          // persistent waves: ~8 tiles/wave, amortize B setup
    if (blocks < 1) blocks = 1;

    kron_linear_wmma_f32<<<blocks, 256, 0, stream>>>(x, f1, f2, b, out, n_rows);
}